// ArgmaxPositions_68513318306404
// MI455X (gfx1250) — compile-verified
//
#include <hip/hip_runtime.h>

// Problem constants (fixed by setup_inputs)
#define HOUT 39
#define WOUT 39
#define NPATCH (HOUT * WOUT)      // 1521 output spatial positions
#define COUT 64
#define WIN  160
#define CIN  4
#define FS   8
#define STRD 4
#define K_ELEMS (FS * FS * CIN)   // 256 elements per patch / filter
#define WFLOATS (K_ELEMS * COUT)  // 16384 weight floats = 64 KB
#define PPB 4                     // patches per 256-thread block

// ---------------------------------------------------------------------------
// CDNA5 async global->LDS path (GLOBAL_LOAD_ASYNC_TO_LDS_* + S_WAIT_ASYNCCNT)
// ---------------------------------------------------------------------------
#if __has_builtin(__builtin_amdgcn_global_load_async_to_lds_b32) && \
    __has_builtin(__builtin_amdgcn_s_wait_asynccnt)
#define HAVE_ASYNC 1
#else
#define HAVE_ASYNC 0
#endif
#if HAVE_ASYNC && __has_builtin(__builtin_amdgcn_global_load_async_to_lds_b128)
#define HAVE_ASYNC128 1
#else
#define HAVE_ASYNC128 0
#endif

typedef int v4i __attribute__((vector_size(16)));
typedef __attribute__((address_space(3))) int lds_i32_t;
typedef __attribute__((address_space(1))) int gbl_i32_t;
typedef __attribute__((address_space(3))) v4i lds_v4i_t;
typedef __attribute__((address_space(1))) v4i gbl_v4i_t;

// Copy 16 bytes (4 floats, both sides 16B-aligned) global -> LDS.
__device__ __forceinline__ void copy16_to_lds(const float* g, float* l) {
#if HAVE_ASYNC128
    __builtin_amdgcn_global_load_async_to_lds_b128((gbl_v4i_t*)g, (lds_v4i_t*)l, 0, 0);
#elif HAVE_ASYNC
    __builtin_amdgcn_global_load_async_to_lds_b32((gbl_i32_t*)(g + 0), (lds_i32_t*)(l + 0), 0, 0);
    __builtin_amdgcn_global_load_async_to_lds_b32((gbl_i32_t*)(g + 1), (lds_i32_t*)(l + 1), 0, 0);
    __builtin_amdgcn_global_load_async_to_lds_b32((gbl_i32_t*)(g + 2), (lds_i32_t*)(l + 2), 0, 0);
    __builtin_amdgcn_global_load_async_to_lds_b32((gbl_i32_t*)(g + 3), (lds_i32_t*)(l + 3), 0, 0);
#else
    l[0] = g[0]; l[1] = g[1]; l[2] = g[2]; l[3] = g[3];
#endif
}

// 256 threads = 8 wave32s per block; 4 patches per block, thread (sub,c) owns
// output channel c of patch slot sub. Whole weight tensor (64 KB) + 4 patches
// (4 KB) staged into LDS with async copies; inner scan is pure LDS.
__global__ void __launch_bounds__(256)
argmax_scatter_kernel(const float* __restrict__ rel,  // [39,39,64] relevance
                      const float* __restrict__ x,    // [160,160,4] input map
                      const float* __restrict__ w,    // [8,8,4,64] = [k=256][c=64]
                      float* __restrict__ out)        // [160,160,4] accumulator
{
    __shared__ __align__(16) float wlds[WFLOATS];        // 64 KB
    __shared__ __align__(16) float plds[PPB * K_ELEMS];  // 4 KB

    const int tid = threadIdx.x;
    const int sub = tid >> 6;     // patch slot 0..3
    const int c   = tid & 63;     // output channel

    // Stage full weight tensor: 16 rounds x 256 lanes x 16 B (all coalesced).
#pragma unroll
    for (int it = 0; it < WFLOATS / (256 * 4); ++it) {
        const int e = (it * 256 + tid) * 4;
        copy16_to_lds(w + e, wlds + e);
    }

    // Stage this slot's 8x8x4 patch: chunk c covers floats [4c, 4c+4) in
    // k-order; k = px*32 + (py*4+pc) with the 32-float row contiguous in x.
    const int p0 = blockIdx.x * PPB + sub;
    const bool valid = (p0 < NPATCH);
    int i = 0, j = 0, gx0 = 0, gy0 = 0;
    if (valid) {
        i = p0 / WOUT;
        j = p0 - i * WOUT;
        gx0 = STRD * i;
        gy0 = STRD * j;
        const int px = c >> 3;
        const int fo = (c & 7) * 4;   // float offset within the 32-float row
        const float* gp = x + ((gx0 + px) * WIN + gy0) * CIN + fo;  // 16B-aligned
        copy16_to_lds(gp, plds + sub * K_ELEMS + c * 4);
    }
#if HAVE_ASYNC
    __builtin_amdgcn_s_wait_asynccnt(0);   // this wave's async copies complete
#endif
    __syncthreads();                        // all 8 waves' LDS writes visible

    if (valid) {
        // First-occurrence argmax over k of patch[k]*w[k][c] (strict > == jnp.argmax).
        const float* patch = plds + sub * K_ELEMS;  // broadcast reads (no conflicts)
        float best  = -__builtin_inff();
        int   bestk = 0;
#pragma unroll 8
        for (int k = 0; k < K_ELEMS; ++k) {
            const float v = patch[k] * wlds[k * COUT + c];  // lanes -> 32 distinct banks
            if (v > best) { best = v; bestk = k; }
        }

        // Decompose flat patch index and scatter-add relevance.
        const int pc = bestk & 3;
        const int py = (bestk >> 2) & 7;
        const int px = bestk >> 5;
        const int flat = ((gx0 + px) * WIN + (gy0 + py)) * CIN + pc;
        const float r = rel[(i * WOUT + j) * COUT + c];
        atomicAdd(out + flat, r);   // global_atomic_add_f32 (no return)
    }
}

extern "C" void kernel_launch(void* const* d_in, const int* in_sizes, int n_in,
                              void* d_out, int out_size, void* d_ws, size_t ws_size,
                              hipStream_t stream) {
    const float* rel = (const float*)d_in[0];   // [1,39,39,64]
    const float* x   = (const float*)d_in[1];   // [1,160,160,4]
    const float* w   = (const float*)d_in[2];   // [8,8,4,64]
    // d_in[3] = stride (4), d_in[4] = filter_size (8) -- fixed, baked in.
    float* out = (float*)d_out;                 // [1,160,160,4] = 102400 floats

    (void)hipMemsetAsync(out, 0, (size_t)out_size * sizeof(float), stream);

    const int nblocks = (NPATCH + PPB - 1) / PPB;   // 381
    argmax_scatter_kernel<<<nblocks, 256, 0, stream>>>(rel, x, w, out);
}